// UpBlock_22737556865527
// MI455X (gfx1250) — compile-verified
//
#include <hip/hip_runtime.h>
#include <math.h>

typedef float v2f __attribute__((ext_vector_type(2)));
typedef float v8f __attribute__((ext_vector_type(8)));

#define EPSF 1e-5f

enum { ACT_NONE = 0, ACT_BN_RELU = 1, ACT_GELU = 2, ACT_SOFTPLUS = 3 };

// ---------------------------------------------------------------------------
// fp32 WMMA GEMM (V_WMMA_F32_16X16X4_F32).
// Per batch z: OUT(M,S) (+)= W(M,Kd) @ X(Kd,S)  [+bias][BN+ReLU | GELU | softplus]
// Requires M%16==0, Kd%4==0, S%64==0 (true for every GEMM in this net).
// One wave computes a 16x64 output tile (4 accumulators, A reused 4x).
// ---------------------------------------------------------------------------
__global__ __launch_bounds__(128)
void gemm_wmma_kernel(float* __restrict__ out, const float* __restrict__ X,
                      const float* __restrict__ W, const float* __restrict__ bias,
                      const float* __restrict__ bn_g, const float* __restrict__ bn_b,
                      const float* __restrict__ bn_m, const float* __restrict__ bn_v,
                      int M, int Kd, int S, int wrs,
                      long x_bs, long out_bs, int act, int accum)
{
  const int lane  = threadIdx.x;                 // 0..31 (wave32)
  const int mtile = blockIdx.y * 4 + threadIdx.y;
  if (mtile * 16 >= M) return;
  const int m0 = mtile * 16;
  const int n0 = blockIdx.x * 64;
  const float* Xb = X + (long)blockIdx.z * x_bs;
  float*       Ob = out + (long)blockIdx.z * out_bs;

  const int half = lane >> 4;   // 0: lanes 0-15, 1: lanes 16-31
  const int lr   = lane & 15;
  const float* Wrow = W + (long)(m0 + lr) * wrs;

  v8f a0 = {}, a1 = {}, a2 = {}, a3 = {};
  for (int k0 = 0; k0 < Kd; k0 += 4) {
    // A tile 16x4: lanes0-15 row M=lr K={k0,k0+1}; lanes16-31 K={k0+2,k0+3}
    v2f av;
    av.x = Wrow[k0 + 2 * half];
    av.y = Wrow[k0 + 2 * half + 1];
    // B tile 4x16: row-per-16-lanes, K striped {0,2}/{1,3} mirroring A
    const float* x0 = Xb + (long)(k0 + 2 * half) * S + n0 + lr;
    v2f b0, b1, b2, b3;
    b0.x = x0[0];  b0.y = x0[S];
    b1.x = x0[16]; b1.y = x0[S + 16];
    b2.x = x0[32]; b2.y = x0[S + 32];
    b3.x = x0[48]; b3.y = x0[S + 48];
    a0 = __builtin_amdgcn_wmma_f32_16x16x4_f32(false, av, false, b0, (short)0, a0, false, false);
    a1 = __builtin_amdgcn_wmma_f32_16x16x4_f32(false, av, false, b1, (short)0, a1, false, false);
    a2 = __builtin_amdgcn_wmma_f32_16x16x4_f32(false, av, false, b2, (short)0, a2, false, false);
    a3 = __builtin_amdgcn_wmma_f32_16x16x4_f32(false, av, false, b3, (short)0, a3, false, false);
  }

  v8f accs[4] = {a0, a1, a2, a3};
  #pragma unroll
  for (int j = 0; j < 4; ++j) {
    const int col = n0 + j * 16 + lr;
    #pragma unroll
    for (int r = 0; r < 8; ++r) {
      const int m = m0 + r + 8 * half;   // C/D layout: VGPR r -> M=r (+8 for hi half)
      float v = accs[j][r];
      if (bias) v += bias[m];
      if (act == ACT_BN_RELU) {
        v = (v - bn_m[m]) * rsqrtf(bn_v[m] + EPSF) * bn_g[m] + bn_b[m];
        v = fmaxf(v, 0.0f);
      } else if (act == ACT_GELU) {
        v = 0.5f * v * (1.0f + erff(v * 0.70710678118654752f));
      } else if (act == ACT_SOFTPLUS) {
        v = (v > 20.0f) ? v : log1pf(expf(v));
      }
      const long o = (long)m * S + col;
      if (accum) v += Ob[o];
      Ob[o] = v;
    }
  }
}

// ---------------------------------------------------------------------------
// Depthwise conv (zero pad), optional bias / SiLU / add-input / accumulate-out
// mode: 0 = write, 1 = write + add input (residual), 2 = accumulate into out
// ---------------------------------------------------------------------------
__global__ __launch_bounds__(256)
void dwconv_kernel(float* __restrict__ out, const float* __restrict__ in,
                   const float* __restrict__ w, const float* __restrict__ bias,
                   int C, int H, int W, int ks, int mode, int act, long N)
{
  const long idx = (long)blockIdx.x * 256 + threadIdx.x;
  if (idx >= N) return;
  const int x = (int)(idx % W);
  long t = idx / W;
  const int y = (int)(t % H); t /= H;
  const int c = (int)(t % C);
  const float* ib = in + t * (long)H * W;
  const float* wc = w + (long)c * ks * ks;
  const int pad = ks >> 1;
  float acc = 0.0f;
  for (int i = 0; i < ks; ++i) {
    const int yy = y + i - pad;
    if (yy < 0 || yy >= H) continue;
    for (int j = 0; j < ks; ++j) {
      const int xx = x + j - pad;
      if (xx < 0 || xx >= W) continue;
      acc += ib[(long)yy * W + xx] * wc[i * ks + j];
    }
  }
  if (bias) acc += bias[c];
  if (act == 1) acc = acc / (1.0f + expf(-acc));   // SiLU
  if (mode == 1) acc += in[idx];
  else if (mode == 2) acc += out[idx];
  out[idx] = acc;
}

// ---------------------------------------------------------------------------
// LayerNorm over channel dim (C=256) at each (b, spatial) of an NCHW tensor
// ---------------------------------------------------------------------------
__global__ __launch_bounds__(256)
void ln_ch_kernel(float* __restrict__ out, const float* __restrict__ in,
                  const float* __restrict__ g, const float* __restrict__ bt, int S)
{
  __shared__ float s1[256];
  __shared__ float s2[256];
  const int c = threadIdx.x;
  const int s = blockIdx.x;
  const int b = blockIdx.y;
  const long off = ((long)b * 256 + c) * S + s;
  const float v = in[off];
  s1[c] = v; s2[c] = v * v;
  __syncthreads();
  for (int o = 128; o > 0; o >>= 1) {
    if (c < o) { s1[c] += s1[c + o]; s2[c] += s2[c + o]; }
    __syncthreads();
  }
  const float mu  = s1[0] * (1.0f / 256.0f);
  const float var = s2[0] * (1.0f / 256.0f) - mu * mu;
  out[off] = (v - mu) * rsqrtf(var + EPSF) * g[c] + bt[c];
}

// ---------------------------------------------------------------------------
// LKPE pixel-shuffle (B,1024,32,32)->(B,64,64,256) + LN(last dim) -> NCHW out
// out[b,d,2h+a,2w+cb] from in[b,(a*2+cb)*256+d,h,w]
// ---------------------------------------------------------------------------
__global__ __launch_bounds__(256)
void shuffle_ln_kernel(float* __restrict__ out, const float* __restrict__ in,
                       const float* __restrict__ g, const float* __restrict__ bt)
{
  __shared__ float s1[256];
  __shared__ float s2[256];
  const int d  = threadIdx.x;
  const int sp = blockIdx.x;             // y*64 + x
  const int b  = blockIdx.y;
  const int xq = sp & 63, y = sp >> 6;
  const int h = y >> 1, a = y & 1, w = xq >> 1, cb = xq & 1;
  const int ch = (a * 2 + cb) * 256 + d;
  const float v = in[((long)b * 1024 + ch) * 1024 + h * 32 + w];
  s1[d] = v; s2[d] = v * v;
  __syncthreads();
  for (int o = 128; o > 0; o >>= 1) {
    if (d < o) { s1[d] += s1[d + o]; s2[d] += s2[d + o]; }
    __syncthreads();
  }
  const float mu  = s1[0] * (1.0f / 256.0f);
  const float var = s2[0] * (1.0f / 256.0f) - mu * mu;
  out[((long)b * 256 + d) * 4096 + sp] = (v - mu) * rsqrtf(var + EPSF) * g[d] + bt[d];
}

// scan-direction index map: k0=fwd, k1=transposed, k2=rev, k3=rev-transposed
__device__ __forceinline__ int scan_map(int k, int l) {
  const int ll = (k & 2) ? (4095 - l) : l;
  return (k & 1) ? ((ll & 63) * 64 + (ll >> 6)) : ll;
}

// ---------------------------------------------------------------------------
// x_dbl[b,k,c,l] = sum_d x[b,d,map_k(l)] * xproj_w[k,c,d]   (c = 0..17)
// Weights staged in LDS (18x256 = 18KB); one thread per l.
// ---------------------------------------------------------------------------
__global__ __launch_bounds__(256)
void xproj_kernel(float* __restrict__ xdbl, const float* __restrict__ xc,
                  const float* __restrict__ xw)
{
  __shared__ float ws[18 * 256];
  const int tid = threadIdx.x;
  const int bk = blockIdx.y;
  const int b = bk >> 2, k = bk & 3;
  for (int e = tid; e < 18 * 256; e += 256) ws[e] = xw[k * 18 * 256 + e];
  __syncthreads();
  const int l = blockIdx.x * 256 + tid;
  const int pos = scan_map(k, l);
  float acc[18];
  #pragma unroll
  for (int c = 0; c < 18; ++c) acc[c] = 0.0f;
  const float* xb = xc + (long)b * 256 * 4096 + pos;
  for (int d = 0; d < 256; ++d) {
    const float xv = xb[(long)d * 4096];
    #pragma unroll
    for (int c = 0; c < 18; ++c) acc[c] += xv * ws[c * 256 + d];
  }
  float* ob = xdbl + (long)bk * 18 * 4096 + l;
  #pragma unroll
  for (int c = 0; c < 18; ++c) ob[(long)c * 4096] = acc[c];
}

// ---------------------------------------------------------------------------
// Selective scan (N=1): per (b,k,d) thread, h = exp(dt*A)*h + dt*B*u ; y=h*C+D*u
// dt already softplus'ed (fused into dtproj GEMM epilogue).
// ---------------------------------------------------------------------------
__global__ __launch_bounds__(256)
void scan_kernel(float* __restrict__ ys, const float* __restrict__ dts,
                 const float* __restrict__ xdbl, const float* __restrict__ xc,
                 const float* __restrict__ Alog, const float* __restrict__ Ds)
{
  const int d = threadIdx.x;
  const int bk = blockIdx.x;
  const int b = bk >> 2, k = bk & 3;
  const int kd = k * 256 + d;
  const float Av = -expf(Alog[kd]);           // A = -exp(A_log), N=1
  const float Dv = Ds[kd];
  const float* dtrow = dts + ((long)bk * 256 + d) * 4096;
  const float* Bs = xdbl + ((long)bk * 18 + 16) * 4096;  // c=16 row (uniform bcast)
  const float* Cs = Bs + 4096;                           // c=17 row
  const float* xrow = xc + ((long)b * 256 + d) * 4096;
  float* yrow = ys + ((long)bk * 256 + d) * 4096;
  float h = 0.0f;
  for (int l = 0; l < 4096; ++l) {
    const float dt = dtrow[l];
    const float u = xrow[scan_map(k, l)];
    h = expf(dt * Av) * h + dt * Bs[l] * u;
    yrow[l] = h * Cs[l] + Dv * u;
  }
}

// ---------------------------------------------------------------------------
// Merge 4 scan directions: y[b,d,h,w] = ys0[s] + ys2[L-1-s] + ys1[st] + ys3[L-1-st]
// ---------------------------------------------------------------------------
__global__ __launch_bounds__(256)
void combine_kernel(float* __restrict__ y, const float* __restrict__ ys)
{
  const long idx = (long)blockIdx.x * 256 + threadIdx.x;   // over B*256*4096
  const int s = (int)(idx & 4095);
  const long t = idx >> 12;
  const int d = (int)(t & 255);
  const int b = (int)(t >> 8);
  const int hh = s >> 6, ww = s & 63;
  const int st = ww * 64 + hh;
  const float* base = ys + ((long)b * 4 * 256 + d) * 4096;
  const long ksd = 256L * 4096;
  y[idx] = base[s] + base[2 * ksd + (4095 - s)]
         + base[ksd + st] + base[3 * ksd + (4095 - st)];
}

// ---------------------------------------------------------------------------
// Host-side launch helpers + full pipeline
// ---------------------------------------------------------------------------
static inline void gemm(hipStream_t st, float* out, const float* X, const float* W,
                        const float* bias, const float* g, const float* b,
                        const float* m, const float* v,
                        int M, int Kd, int S, int wrs, long xbs, long obs,
                        int act, int accum)
{
  dim3 grid(S / 64, (M + 63) / 64, 8);
  gemm_wmma_kernel<<<grid, dim3(32, 4), 0, st>>>(out, X, W, bias, g, b, m, v,
                                                 M, Kd, S, wrs, xbs, obs, act, accum);
}

static inline void dwc(hipStream_t st, float* out, const float* in, const float* w,
                       const float* bias, int C, int H, int W, int ks, int mode, int act)
{
  const long N = 8L * C * H * W;
  dwconv_kernel<<<dim3((unsigned)((N + 255) / 256)), dim3(256), 0, st>>>(
      out, in, w, bias, C, H, W, ks, mode, act, N);
}

extern "C" void kernel_launch(void* const* d_in, const int* in_sizes, int n_in,
                              void* d_out, int out_size, void* d_ws, size_t ws_size,
                              hipStream_t stream)
{
  (void)in_sizes; (void)n_in; (void)out_size; (void)ws_size;
  const float* inp  = (const float*)d_in[0];   // (8,512,32,32)
  const float* skip = (const float*)d_in[1];   // (8,256,64,64)
  auto P = [&](int i) { return (const float*)d_in[i]; };

  float* ws   = (float*)d_ws;
  float* r0   = ws;                          // 8,388,608 f (32 MB)
  float* r1   = ws + (size_t)8388608;        // 32 MB
  float* r2   = ws + (size_t)16777216;       // 32 MB
  float* big0 = ws + (size_t)25165824;       // 33.5M f: dts, then fc1 z
  float* big1 = ws + (size_t)58720256;       // 33.5M f: ys,  then ms-out
  float* xdbl = ws + (size_t)92274688;       // 2.36M f
  float* x    = (float*)d_out;               // (8,256,64,64) — running residual

  // ---- LKPE: conv1x1(512->1024)+bias+BN+ReLU -> dw3x3+bias -> shuffle+LN ----
  gemm(stream, r0, inp, P(2), P(3), P(4), P(5), P(6), P(7),
       1024, 512, 1024, 512, 512L * 1024, 1024L * 1024, ACT_BN_RELU, 0);
  dwc(stream, r1, r0, P(8), P(9), 1024, 32, 32, 3, 0, 0);
  shuffle_ln_kernel<<<dim3(4096, 8), 256, 0, stream>>>(r2, r1, P(10), P(11));

  // ---- concat conv: x = cat_w[:, :256]@lkpe + cat_w[:, 256:]@skip + b ----
  gemm(stream, x, r2,   P(12),       nullptr, nullptr, nullptr, nullptr, nullptr,
       256, 256, 4096, 512, 256L * 4096, 256L * 4096, ACT_NONE, 0);
  gemm(stream, x, skip, P(12) + 256, P(13),   nullptr, nullptr, nullptr, nullptr,
       256, 256, 4096, 512, 256L * 4096, 256L * 4096, ACT_NONE, 1);

  // ---- 2 x (SS2D block + multiscale MLP) ----
  for (int i = 0; i < 2; ++i) {
    const int bi = 14 + 23 * i;
    // ln1
    ln_ch_kernel<<<dim3(4096, 8), 256, 0, stream>>>(r2, x, P(bi + 0), P(bi + 1), 4096);
    // in conv (256->256) + bias
    gemm(stream, r0, r2, P(bi + 2), P(bi + 3), nullptr, nullptr, nullptr, nullptr,
         256, 256, 4096, 256, 256L * 4096, 256L * 4096, ACT_NONE, 0);
    // dw3x3 + SiLU
    dwc(stream, r1, r0, P(bi + 4), nullptr, 256, 64, 64, 3, 0, 1);
    // x_dbl = xproj over 4 scan directions (gathers fused, no xs materialization)
    xproj_kernel<<<dim3(16, 32), 256, 0, stream>>>(xdbl, r1, P(bi + 5));
    // dt = softplus(dtproj @ x_dbl[:,:16] + dt_bias)  — WMMA GEMM, K=16
    for (int k = 0; k < 4; ++k) {
      gemm(stream, big0 + (size_t)k * 256 * 4096, xdbl + (size_t)k * 18 * 4096,
           P(bi + 6) + (size_t)k * 256 * 16, P(bi + 7) + (size_t)k * 256,
           nullptr, nullptr, nullptr, nullptr,
           256, 16, 4096, 16, 4L * 18 * 4096, 4L * 256 * 4096, ACT_SOFTPLUS, 0);
    }
    // selective scan (+ Ds*u fused)
    scan_kernel<<<dim3(32), 256, 0, stream>>>(big1, big0, xdbl, r1, P(bi + 8), P(bi + 9));
    // merge 4 directions
    combine_kernel<<<dim3(32768), 256, 0, stream>>>(r0, big1);
    // out LN
    ln_ch_kernel<<<dim3(4096, 8), 256, 0, stream>>>(r2, r0, P(bi + 10), P(bi + 11), 4096);
    // out conv + bias, residual-accumulate into x
    gemm(stream, x, r2, P(bi + 12), P(bi + 13), nullptr, nullptr, nullptr, nullptr,
         256, 256, 4096, 256, 256L * 4096, 256L * 4096, ACT_NONE, 1);
    // ln2
    ln_ch_kernel<<<dim3(4096, 8), 256, 0, stream>>>(r2, x, P(bi + 14), P(bi + 15), 4096);
    // fc1 (256->1024) + bias + exact GELU
    gemm(stream, big0, r2, P(bi + 16), P(bi + 17), nullptr, nullptr, nullptr, nullptr,
         1024, 256, 4096, 256, 256L * 4096, 1024L * 4096, ACT_GELU, 0);
    // z + dw1 + dw3 + dw5 (read big0, build into big1)
    dwc(stream, big1, big0, P(bi + 18), nullptr, 1024, 64, 64, 1, 1, 0);
    dwc(stream, big1, big0, P(bi + 19), nullptr, 1024, 64, 64, 3, 2, 0);
    dwc(stream, big1, big0, P(bi + 20), nullptr, 1024, 64, 64, 5, 2, 0);
    // fc2 (1024->256) + bias, residual-accumulate into x
    gemm(stream, x, big1, P(bi + 21), P(bi + 22), nullptr, nullptr, nullptr, nullptr,
         256, 1024, 4096, 1024, 1024L * 4096, 256L * 4096, ACT_NONE, 1);
  }
}